// MSWSA_67894843015209
// MI455X (gfx1250) — compile-verified
//
#include <hip/hip_runtime.h>
#include <hip/hip_bf16.h>

typedef float v2f __attribute__((ext_vector_type(2)));
typedef float v8f __attribute__((ext_vector_type(8)));

#define WMMA_F32(a, b, c) __builtin_amdgcn_wmma_f32_16x16x4_f32( \
    false, (a), false, (b), (short)0, (c), false, false)

// Async DMA: copy 16 bytes per lane from global memory straight into LDS.
// VDST = per-lane LDS byte offset (wave-relative), VADDR = 64-bit global addr.
// Tracked by ASYNCcnt (CDNA5 ISA).
__device__ __forceinline__ void async_copy_b128(unsigned lds_off, const void* gptr) {
    asm volatile("global_load_async_to_lds_b128 %0, %1, off"
                 :: "v"(lds_off), "v"((unsigned long long)(uintptr_t)gptr)
                 : "memory");
}
__device__ __forceinline__ void wait_async0() {
    asm volatile("s_wait_asynccnt 0x0" ::: "memory");
}

// ---------------------------------------------------------------------------
// fp32 WMMA GEMM:  C[M,N] = A[M,K] @ B[K,N] + bias[N]
// Block tile 256x64, 256 threads = 8 waves; wave owns 32 rows x 64 cols
// (2 row-frags x 4 col-frags = 8 fp32-WMMA accumulators).
// A tile prefetched with GLOBAL_LOAD_ASYNC_TO_LDS_B128 (no staging VGPRs,
// no spills); B tile register-staged (1 float4) and transposed into LDS so
// every fragment read is one contiguous ds_load_b64. Double-buffered LDS,
// 1 barrier per K-chunk. Requires M%256==0, N%64==0, K%16==0.
// ---------------------------------------------------------------------------
__global__ __launch_bounds__(256, 2) void MSWSA_gemm_bias(
    const float* __restrict__ A, const float* __restrict__ B,
    const float* __restrict__ bias, float* __restrict__ C,
    int M, int N, int K)
{
    __shared__ __attribute__((aligned(16))) float As[2][256][20];
    __shared__ __attribute__((aligned(16))) float BsT[2][64][20];

    const int tid  = threadIdx.x;
    const int lane = tid & 31;
    const int wave = tid >> 5;                 // 0..7
    const int bm   = blockIdx.x * 256;
    const int bn   = blockIdx.y * 64;

    const int frag_row = lane & 15;            // A row / B col within frag
    const int frag_k   = (lane >> 4) * 2;      // k-pair base: 0 or 2
    const int crow_off = 8 * (lane >> 4);      // C/D row offset
    const int ccol     = lane & 15;            // C/D column

    // Per-thread async-copy slots for the A tile (256x16 fl = 1024 x 16B).
    const float* agp[4];
    unsigned     alds[4];
    #pragma unroll
    for (int i = 0; i < 4; ++i) {
        int v   = tid + i * 256;               // 0..1023
        int row = v >> 2;                      // 4 x 16B per 16-wide row
        int c4  = v & 3;
        agp[i]  = A + (size_t)(bm + row) * K + c4 * 4;
        alds[i] = (unsigned)(uintptr_t)&As[0][row][c4 * 4];
    }
    const unsigned ABUF = sizeof(float) * 256 * 20;     // buffer stride in LDS

    // B tile: 16x64 fl = 256 float4, one per thread, register-staged.
    const int b_krow = tid >> 4;
    const int b_c4   = tid & 15;
    const float* bgp = B + (size_t)b_krow * N + bn + b_c4 * 4;
    float4 breg;

    auto issue_a = [&](int buf) {
        #pragma unroll
        for (int i = 0; i < 4; ++i) {
            async_copy_b128(alds[i] + (unsigned)buf * ABUF, agp[i]);
            agp[i] += 16;                      // advance one K-chunk
        }
    };
    auto load_b  = [&]() { breg = *(const float4*)bgp; bgp += (size_t)16 * N; };
    auto store_b = [&](int buf) {
        BsT[buf][b_c4 * 4 + 0][b_krow] = breg.x;       // transpose into LDS
        BsT[buf][b_c4 * 4 + 1][b_krow] = breg.y;
        BsT[buf][b_c4 * 4 + 2][b_krow] = breg.z;
        BsT[buf][b_c4 * 4 + 3][b_krow] = breg.w;
    };

    v8f acc[2][4] = {};

    issue_a(0);
    load_b();
    wait_async0();
    store_b(0);
    __syncthreads();

    const int nk = K / 16;
    #pragma unroll 2
    for (int t = 0; t < nk; ++t) {
        const int cur = t & 1;
        if (t + 1 < nk) { issue_a(cur ^ 1); load_b(); }   // prefetch next

        #pragma unroll
        for (int kk = 0; kk < 16; kk += 4) {
            v2f a0 = *(const v2f*)&As[cur][wave * 32 + frag_row][kk + frag_k];
            v2f a1 = *(const v2f*)&As[cur][wave * 32 + 16 + frag_row][kk + frag_k];
            #pragma unroll
            for (int c = 0; c < 4; ++c) {
                v2f b = *(const v2f*)&BsT[cur][c * 16 + frag_row][kk + frag_k];
                acc[0][c] = WMMA_F32(a0, b, acc[0][c]);
                acc[1][c] = WMMA_F32(a1, b, acc[1][c]);
            }
        }
        if (t + 1 < nk) { wait_async0(); store_b(cur ^ 1); }
        __syncthreads();
    }

    // Bias + store.
    #pragma unroll
    for (int c = 0; c < 4; ++c) {
        const int col = bn + c * 16 + ccol;
        const float bv = bias[col];
        #pragma unroll
        for (int rf = 0; rf < 2; ++rf) {
            #pragma unroll
            for (int v = 0; v < 8; ++v) {
                const int row = bm + wave * 32 + rf * 16 + v + crow_off;
                C[(size_t)row * N + col] = acc[rf][c][v] + bv;
            }
        }
    }
}

// ---------------------------------------------------------------------------
// Row-window attention: one block per (b, head, row). 128 threads = 4 waves.
// qkv layout: [B, N, 3, nh, dh] flat = (b*4096+n)*2304 + s*768 + h*64 + d.
// S = (q k^T) * 0.125 ; P = softmax(S) ; O = P v.
// v stored transposed in LDS so P@v B-fragments are contiguous ds_load_b64.
// ---------------------------------------------------------------------------
__global__ __launch_bounds__(128, 2) void MSWSA_attn(
    const float* __restrict__ qkv, float* __restrict__ out)
{
    __shared__ float qs[64][68];    // q, later P       (stride 68: even, 16B)
    __shared__ float ks[64][68];    // k row-major ( == B for q k^T )
    __shared__ float vsT[64][68];   // v transposed: vsT[d][u] = v[u][d]

    const int tid  = threadIdx.x;
    const int lane = tid & 31;
    const int wave = tid >> 5;                 // 0..3
    const int r = blockIdx.x & 63;             // row in grid
    const int h = (blockIdx.x >> 6) % 12;      // head
    const int b = blockIdx.x / (64 * 12);      // batch

    const size_t base = ((size_t)(b * 4096 + r * 64)) * 2304 + h * 64;

    // Stage q, k, v (each 64x64 = 1024 float4; 8 float4 per thread each).
    #pragma unroll
    for (int i = 0; i < 8; ++i) {
        int v   = tid + i * 128;               // 0..1023
        int row = v >> 4;                      // token index u
        int c4  = v & 15;
        const float* src = qkv + base + (size_t)row * 2304;
        float4 dq = *(const float4*)(src + c4 * 4);
        float4 dk = *(const float4*)(src + 768 + c4 * 4);
        float4 dv = *(const float4*)(src + 1536 + c4 * 4);
        *(float4*)&qs[row][c4 * 4] = dq;
        *(float4*)&ks[row][c4 * 4] = dk;
        vsT[c4 * 4 + 0][row] = dv.x;           // transpose v into LDS
        vsT[c4 * 4 + 1][row] = dv.y;
        vsT[c4 * 4 + 2][row] = dv.z;
        vsT[c4 * 4 + 3][row] = dv.w;
    }
    __syncthreads();

    const int frag_row = lane & 15;
    const int frag_k   = (lane >> 4) * 2;
    const int crow_off = 8 * (lane >> 4);
    const int ccol     = lane & 15;

    // S = q @ k^T : wave owns rows [16*wave, 16*wave+16), all 64 columns.
    v8f sacc[4] = {};
    #pragma unroll
    for (int kk = 0; kk < 64; kk += 4) {
        v2f a = *(const v2f*)&qs[wave * 16 + frag_row][kk + frag_k];
        #pragma unroll
        for (int c = 0; c < 4; ++c) {
            v2f bb = *(const v2f*)&ks[c * 16 + frag_row][kk + frag_k];
            sacc[c] = WMMA_F32(a, bb, sacc[c]);
        }
    }
    __syncthreads();                           // all waves done reading qs

    // Write scaled scores into qs (reuse as S/P buffer).
    #pragma unroll
    for (int c = 0; c < 4; ++c)
        #pragma unroll
        for (int v = 0; v < 8; ++v)
            qs[wave * 16 + v + crow_off][c * 16 + ccol] = sacc[c][v] * 0.125f;
    __syncthreads();

    // Row softmax: thread t handles row t (fp32 exact, v_exp_f32).
    if (tid < 64) {
        float m = -3.402823e38f;
        for (int j = 0; j < 64; ++j) m = fmaxf(m, qs[tid][j]);
        float s = 0.f;
        for (int j = 0; j < 64; ++j) {
            float e = __expf(qs[tid][j] - m);
            qs[tid][j] = e;
            s += e;
        }
        float inv = 1.0f / s;
        for (int j = 0; j < 64; ++j) qs[tid][j] *= inv;
    }
    __syncthreads();

    // O = P @ v : fragment pair = vsT[d][u..u+1] contiguous.
    v8f oacc[4] = {};
    #pragma unroll
    for (int kk = 0; kk < 64; kk += 4) {
        v2f a = *(const v2f*)&qs[wave * 16 + frag_row][kk + frag_k];
        #pragma unroll
        for (int c = 0; c < 4; ++c) {
            v2f bb = *(const v2f*)&vsT[c * 16 + frag_row][kk + frag_k];
            oacc[c] = WMMA_F32(a, bb, oacc[c]);
        }
    }

    // Store to attention output (B, N, C) with C-offset h*64.
    #pragma unroll
    for (int c = 0; c < 4; ++c) {
        #pragma unroll
        for (int v = 0; v < 8; ++v) {
            const int tok = r * 64 + wave * 16 + v + crow_off;
            out[((size_t)(b * 4096 + tok)) * 768 + h * 64 + c * 16 + ccol] = oacc[c][v];
        }
    }
}

extern "C" void kernel_launch(void* const* d_in, const int* in_sizes, int n_in,
                              void* d_out, int out_size, void* d_ws, size_t ws_size,
                              hipStream_t stream) {
    (void)in_sizes; (void)n_in; (void)out_size; (void)ws_size;
    // Inputs: 0:x  1:H  2:W  3:w_qkv  4:b_qkv  5:w_proj  6:b_proj
    const float* x      = (const float*)d_in[0];
    const float* w_qkv  = (const float*)d_in[3];
    const float* b_qkv  = (const float*)d_in[4];
    const float* w_proj = (const float*)d_in[5];
    const float* b_proj = (const float*)d_in[6];
    float* out = (float*)d_out;

    const int M = 8 * 4096;        // 32768 tokens
    const int C = 768;

    float* qkv_ws  = (float*)d_ws;                       // 32768 x 2304
    float* attn_ws = qkv_ws + (size_t)M * 3 * C;         // 32768 x 768

    // 1) QKV projection: (32768 x 768) @ (768 x 2304) + bias
    MSWSA_gemm_bias<<<dim3(M / 256, (3 * C) / 64), 256, 0, stream>>>(
        x, w_qkv, b_qkv, qkv_ws, M, 3 * C, C);

    // 2) Per-(batch, head, row) windowed attention: 8*12*64 = 6144 blocks
    MSWSA_attn<<<8 * 12 * 64, 128, 0, stream>>>(qkv_ws, attn_ws);

    // 3) Output projection: (32768 x 768) @ (768 x 768) + bias
    MSWSA_gemm_bias<<<dim3(M / 256, C / 64), 256, 0, stream>>>(
        attn_ws, w_proj, b_proj, out, M, C, C);
}